// ModelNew_2473901162644
// MI455X (gfx1250) — compile-verified
//
#include <hip/hip_runtime.h>

// ---------------------------------------------------------------------------
// out[b] = 0.5 * scale * dot(x[b,:], wsum)   where wsum = column-sum(weight)
// B=16384, I=4096, H=1024, all f32. Bandwidth-bound (~272 MiB @ 23.3 TB/s).
// Main kernel uses V_WMMA_F32_16X16X4_F32: one wave owns a 16-row tile of x,
// streams it once from HBM (b64/lane loads matching the A-matrix layout),
// and pulls the broadcast wsum operand from LDS.
// ---------------------------------------------------------------------------

typedef __attribute__((ext_vector_type(2))) float v2f;
typedef __attribute__((ext_vector_type(8))) float v8f;

#define I_DIM 4096
#define H_DIM 1024
#define CHUNKS 8   // deterministic 2-stage weight reduction (no float atomics)
// d_ws layout: [0, CHUNKS*I_DIM) partial row-chunk sums, then [.., +I_DIM) wsum

// Stage 1: each (col, chunk) sums 128 rows of weight -> partials (deterministic)
__global__ void weight_partial_sum(const float* __restrict__ w,
                                   float* __restrict__ part) {
    int col   = blockIdx.x * blockDim.x + threadIdx.x;   // coalesced across lanes
    int chunk = blockIdx.y;
    int h0    = chunk * (H_DIM / CHUNKS);
    float s = 0.f;
#pragma unroll 4
    for (int h = 0; h < H_DIM / CHUNKS; ++h)
        s += w[(size_t)(h0 + h) * I_DIM + col];
    part[(size_t)chunk * I_DIM + col] = s;
}

// Stage 2: fold the CHUNKS partials into wsum (fixed order -> deterministic)
__global__ void weight_final_sum(const float* __restrict__ part,
                                 float* __restrict__ wsum) {
    int col = blockIdx.x * blockDim.x + threadIdx.x;
    float s = 0.f;
#pragma unroll
    for (int c = 0; c < CHUNKS; ++c)
        s += part[(size_t)c * I_DIM + col];
    wsum[col] = s;
}

// Main GEMV via WMMA f32 16x16x4.
// Block = 128 threads = 4 waves; each wave owns a 16-row tile -> 64 rows/block.
// A-matrix layout (ISA 7.12.2, 32-bit A 16x4):
//   lane L (0..15): M=L, VGPR0=K0, VGPR1=K1 ; lane L+16: M=L, VGPR0=K2, VGPR1=K3
// -> per lane, a single float2 load at column k + (lane>>4)*2 builds A exactly.
// B-matrix (4x16, rows striped across lanes in a VGPR): B[k'][n] = wsum[k+k']
// broadcast over n -> the SAME float2 indexing, read from LDS.
__global__ __launch_bounds__(128) void gemv_wmma_f32(
        const float* __restrict__ x,
        const float* __restrict__ wsum,
        const float* __restrict__ scale,
        float* __restrict__ out) {
    __shared__ float sw[I_DIM];  // 16 KiB, reused by all 64 rows of this block

    int tid = threadIdx.x;
#pragma unroll
    for (int i = 0; i < I_DIM / (128 * 4); ++i) {          // 8 iters of float4
        int idx = (i * 128 + tid) * 4;
        *(float4*)&sw[idx] = *(const float4*)&wsum[idx];
    }
    __syncthreads();

    int wave  = tid >> 5;
    int lane  = tid & 31;
    int row16 = lane & 15;
    int koff  = (lane >> 4) * 2;                           // 0 or 2
    size_t rowBase = (size_t)blockIdx.x * 64 + (size_t)wave * 16;
    const float* xrow = x + (rowBase + row16) * I_DIM;

    v8f c0 = {};
    v8f c1 = {};
    for (int k = 0; k < I_DIM; k += 8) {
        v2f a0 = *(const v2f*)(xrow + k + koff);           // global_load_b64
        v2f b0 = *(const v2f*)(&sw[k + koff]);             // ds_load_b64 (broadcast)
        c0 = __builtin_amdgcn_wmma_f32_16x16x4_f32(
                 false, a0, false, b0, (short)0, c0, false, false);
        v2f a1 = *(const v2f*)(xrow + k + 4 + koff);
        v2f b1 = *(const v2f*)(&sw[k + 4 + koff]);
        c1 = __builtin_amdgcn_wmma_f32_16x16x4_f32(
                 false, a1, false, b1, (short)0, c1, false, false);
    }

    float alpha = 0.5f * scale[0];
    // C/D layout: VGPR v, lanes 0-15 -> M=v (N=lane), lanes 16-31 -> M=v+8.
    // Every column N holds the same dot product; lane 0 (N=0) writes rows 0-7,
    // lane 16 (N=0 of upper half) writes rows 8-15.
    if (lane == 0 || lane == 16) {
        size_t r0 = rowBase + (size_t)((lane >> 4) * 8);
#pragma unroll
        for (int v = 0; v < 8; ++v)
            out[r0 + v] = alpha * (c0[v] + c1[v]);
    }
}

extern "C" void kernel_launch(void* const* d_in, const int* in_sizes, int n_in,
                              void* d_out, int out_size, void* d_ws, size_t ws_size,
                              hipStream_t stream) {
    const float* x     = (const float*)d_in[0];
    const float* w     = (const float*)d_in[1];
    const float* scale = (const float*)d_in[2];
    float* out  = (float*)d_out;
    float* part = (float*)d_ws;                 // CHUNKS * I_DIM floats
    float* wsum = part + (size_t)CHUNKS * I_DIM;

    int B = in_sizes[0] / I_DIM;                // 16384

    weight_partial_sum<<<dim3(I_DIM / 256, CHUNKS), 256, 0, stream>>>(w, part);
    weight_final_sum<<<I_DIM / 256, 256, 0, stream>>>(part, wsum);
    gemv_wmma_f32<<<B / 64, 128, 0, stream>>>(x, wsum, scale, out);
}